// Cal_Contrastive_loss_29540785062469
// MI455X (gfx1250) — compile-verified
//
#include <hip/hip_runtime.h>
#include <hip/hip_bf16.h>
#include <math.h>

// Problem constants (match reference)
#define B_SZ   4096
#define D_DIM  128
#define N_SZ   8192         // 2*B
#define NSLOTS 64           // 8 strips * 8 waves -> unique partial-sum slots

typedef __attribute__((ext_vector_type(2))) float v2f;
typedef __attribute__((ext_vector_type(8))) float v8f;

// ---------------------------------------------------------------------------
// Kernel 1: row L2-normalization. One wave (32 lanes) per row; each lane owns
// 4 contiguous floats (32*4 = 128 = D). zn is the workspace copy used by the
// WMMA GEMM. Reference: norm = max(||z||, 1e-8); zn = z / norm.
// ---------------------------------------------------------------------------
__global__ __launch_bounds__(256) void normalize_rows(const float* __restrict__ zi,
                                                      const float* __restrict__ zj,
                                                      float* __restrict__ zn) {
    const int wave = threadIdx.x >> 5;
    const int lane = threadIdx.x & 31;
    const int row  = blockIdx.x * 8 + wave;
    const float* src = (row < B_SZ) ? (zi + (size_t)row * D_DIM)
                                    : (zj + (size_t)(row - B_SZ) * D_DIM);
    float4 v = *(const float4*)(src + lane * 4);
    float s = v.x * v.x + v.y * v.y + v.z * v.z + v.w * v.w;
    // full-wave (32-lane) butterfly reduction
    s += __shfl_xor(s, 1);
    s += __shfl_xor(s, 2);
    s += __shfl_xor(s, 4);
    s += __shfl_xor(s, 8);
    s += __shfl_xor(s, 16);
    const float rn = 1.0f / fmaxf(sqrtf(s), 1e-8f);
    float4 o; o.x = v.x * rn; o.y = v.y * rn; o.z = v.z * rn; o.w = v.w * rn;
    *(float4*)(zn + (size_t)row * D_DIM + lane * 4) = o;
}

// ---------------------------------------------------------------------------
// Kernel 2: fp32 WMMA GEMM (sim = 2 * zn @ zn^T, computed tile-wise, never
// materialized) fused with the exp row-sum and positive-pair extraction.
//
// Fragment layout for V_WMMA_F32_16X16X4_F32 (ISA 7.12.2, wave32):
//   A (16x4):  lanes 0-15 hold M=lane; VGPR pair = (K=2*hk, K=2*hk+1) where
//              hk = lane>>4.  B (4x16) mirrors this with N=lane&15. Since both
//              operands come from the same row-major zn, A and B fragment loads
//              use IDENTICAL addressing: float2 at zn[base+m][4*j + 2*hk].
//   C/D (16x16 f32): VGPR i holds row M = i + 8*hk, column N = lane&15.
//
// Each wave: A-panel (16 rows x K=128) pinned in 64 VGPRs, sweeps 8 of the 512
// column tiles (32 WMMAs each). Epilogue: e = exp(2c-2) (diag zeroed), row
// sums reduced across the 16 lanes of each half-wave via xor shuffles, stored
// to a per-(strip,wave) slot -> fully deterministic (no atomics).
// ---------------------------------------------------------------------------
__global__ __launch_bounds__(256) void simclr_gemm(const float* __restrict__ zn,
                                                   float* __restrict__ partial,
                                                   float* __restrict__ pos) {
    const int rowTile = blockIdx.x;           // 0..511
    const int strip   = blockIdx.y;           // 0..7
    const int wave    = threadIdx.x >> 5;     // 0..7
    const int lane    = threadIdx.x & 31;
    const int m       = lane & 15;
    const int hk      = lane >> 4;            // selects K pair within fragment
    const int rowBase = rowTile * 16;
    const int slot    = strip * 8 + wave;     // 0..63, unique column-tile set

    // Pin the wave's entire A panel (row rowBase+m, all 128 K) in registers:
    // 32 fragments x float2 = 64 VGPRs.
    const float* aRow = zn + (size_t)(rowBase + m) * D_DIM + 2 * hk;
    v2f aa[32];
#pragma unroll
    for (int j = 0; j < 32; ++j)
        aa[j] = *(const v2f*)(aRow + 4 * j);

    float rs[8] = {0.f, 0.f, 0.f, 0.f, 0.f, 0.f, 0.f, 0.f};

    for (int t = 0; t < 8; ++t) {
        const int colTile = slot + t * 64;    // covers all 512 tiles across slots
        const int colBase = colTile * 16;
        const float* bRow = zn + (size_t)(colBase + m) * D_DIM + 2 * hk;
        if (t < 7) // hint next B tile into cache (global_prefetch_b8)
            __builtin_prefetch(zn + (size_t)((slot + (t + 1) * 64) * 16 + m) * D_DIM, 0, 1);

        v8f c = {};
#pragma unroll
        for (int j = 0; j < 32; ++j) {
            v2f bb = *(const v2f*)(bRow + 4 * j);
            c = __builtin_amdgcn_wmma_f32_16x16x4_f32(
                    /*neg_a=*/false, aa[j], /*neg_b=*/false, bb,
                    /*c_mod=*/(short)0, c, /*reuse_a=*/false, /*reuse_b=*/false);
        }

        // Epilogue: sim = 2*dot; bounded in [-2,2] so exp(sim-2) in [~0.018,1]
        // -> no online max needed for a stable logsumexp.
#pragma unroll
        for (int i = 0; i < 8; ++i) {
            const int grow = rowBase + i + 8 * hk;
            const int gcol = colBase + m;
            const float sim = 2.0f * c[i];
            if (gcol == (grow ^ B_SZ))        // partner entry: exactly one writer
                pos[grow] = sim;
            const float e = (gcol == grow) ? 0.0f : __expf(sim - 2.0f);
            rs[i] += e;
        }
    }

    // Sum each row across its 16 columns (xor masks < 16 stay inside each
    // half-wave group, which is exactly the N dimension of the C layout).
#pragma unroll
    for (int i = 0; i < 8; ++i) {
        rs[i] += __shfl_xor(rs[i], 1);
        rs[i] += __shfl_xor(rs[i], 2);
        rs[i] += __shfl_xor(rs[i], 4);
        rs[i] += __shfl_xor(rs[i], 8);
    }
    if (m == 0) {                              // lane 0 -> rows i, lane 16 -> rows i+8
#pragma unroll
        for (int i = 0; i < 8; ++i)
            partial[(size_t)slot * N_SZ + rowBase + i + 8 * hk] = rs[i];
    }
}

// ---------------------------------------------------------------------------
// Kernel 3: deterministic final reduction.
// loss = (1/N) * sum_r [ 2 + log(sum_slots partial[s][r]) - pos[r] ]
// ---------------------------------------------------------------------------
__global__ __launch_bounds__(1024) void final_reduce(const float* __restrict__ partial,
                                                     const float* __restrict__ pos,
                                                     float* __restrict__ out) {
    __shared__ float sdata[1024];
    const int tid = threadIdx.x;
    float s = 0.0f;
    for (int r = tid; r < N_SZ; r += 1024) {
        float rsum = 0.0f;
#pragma unroll 8
        for (int w = 0; w < NSLOTS; ++w)       // fixed order -> deterministic
            rsum += partial[(size_t)w * N_SZ + r];
        s += 2.0f + logf(rsum) - pos[r];
    }
    sdata[tid] = s;
    __syncthreads();
    for (int off = 512; off > 0; off >>= 1) {
        if (tid < off) sdata[tid] += sdata[tid + off];
        __syncthreads();
    }
    if (tid == 0) out[0] = sdata[0] * (1.0f / (float)N_SZ);
}

// ---------------------------------------------------------------------------
extern "C" void kernel_launch(void* const* d_in, const int* in_sizes, int n_in,
                              void* d_out, int out_size, void* d_ws, size_t ws_size,
                              hipStream_t stream) {
    const float* zi = (const float*)d_in[0];   // (B, D) f32
    const float* zj = (const float*)d_in[1];   // (B, D) f32
    float* out = (float*)d_out;                // scalar f32

    float* ws      = (float*)d_ws;
    float* zn      = ws;                                   // N*D      = 4 MB
    float* partial = zn + (size_t)N_SZ * D_DIM;            // 64*N     = 2 MB
    float* pos     = partial + (size_t)NSLOTS * N_SZ;      // N        = 32 KB

    normalize_rows<<<N_SZ / 8, 256, 0, stream>>>(zi, zj, zn);
    simclr_gemm<<<dim3(N_SZ / 16, 8), 256, 0, stream>>>(zn, partial, pos);
    final_reduce<<<1, 1024, 0, stream>>>(partial, pos, out);
}